// MaxPool_63496796504731
// MI455X (gfx1250) — compile-verified
//
#include <hip/hip_runtime.h>
#include <stdint.h>

// Sparse 3D max-pool (kernel==stride==2) as segment-max scatter.
// MI455X / gfx1250: memory-bound (~460 MB total traffic -> ~20us @ 23.3 TB/s).
// Uses CDNA5 GLOBAL_ATOMIC_MAX_NUM_F32 (L2 near-memory float max) for the
// scatter reduction; output (64 MB) is fully L2-resident (192 MB L2).

#define COARSE_RES 64
#define NCHAN      64   // C
#define NQUAD      16   // C/4 float4 quads per point

__global__ __launch_bounds__(256) void mp_init_neginf(float4* __restrict__ out4, int n4) {
    int i = blockIdx.x * blockDim.x + threadIdx.x;
    if (i < n4) {
        float4 v;
        v.x = v.y = v.z = v.w = -__builtin_inff();
        out4[i] = v;
    }
}

__global__ __launch_bounds__(256) void mp_scatter_max(const float* __restrict__ feat,
                                                      const int*   __restrict__ coords,
                                                      float*       __restrict__ out,
                                                      int N) {
    int g = blockIdx.x * blockDim.x + threadIdx.x;
    int n = g >> 4;   // point index  (lanes 0..15 of a half-wave share one point)
    int q = g & 15;   // channel quad (consecutive lanes -> coalesced B128 loads)
    if (n >= N) return;

    // coarse voxel id: floor(ijk/2) linearized as (x*64 + y)*64 + z
    int cx = coords[n * 3 + 0] >> 1;
    int cy = coords[n * 3 + 1] >> 1;
    int cz = coords[n * 3 + 2] >> 1;
    int seg = (cx * COARSE_RES + cy) * COARSE_RES + cz;

    // 16-byte aligned: n*64*4 = 256B, q*16B
    const float4 f = *reinterpret_cast<const float4*>(feat + (size_t)n * NCHAN + q * 4);
    float* p = out + (size_t)seg * NCHAN + q * 4;

    // CDNA5 float-max atomic, non-returning (STOREcnt), device scope so the
    // read-modify-write resolves coherently in L2 across all WGPs.
    // One 64-bit VADDR pair, immediate offsets cover the quad.
    asm volatile(
        "global_atomic_max_num_f32 %0, %1, off scope:SCOPE_DEV\n\t"
        "global_atomic_max_num_f32 %0, %2, off offset:4 scope:SCOPE_DEV\n\t"
        "global_atomic_max_num_f32 %0, %3, off offset:8 scope:SCOPE_DEV\n\t"
        "global_atomic_max_num_f32 %0, %4, off offset:12 scope:SCOPE_DEV"
        :
        : "v"(p), "v"(f.x), "v"(f.y), "v"(f.z), "v"(f.w)
        : "memory");
    // s_endpgm performs an implicit wait-idle, so outstanding atomics drain
    // before the dependent finalize kernel launches (stream ordering).
}

__global__ __launch_bounds__(256) void mp_finalize(float4* __restrict__ out4, int n4) {
    int i = blockIdx.x * blockDim.x + threadIdx.x;
    if (i < n4) {
        float4 v = out4[i];
        v.x = (v.x == -__builtin_inff()) ? 0.0f : v.x;
        v.y = (v.y == -__builtin_inff()) ? 0.0f : v.y;
        v.z = (v.z == -__builtin_inff()) ? 0.0f : v.z;
        v.w = (v.w == -__builtin_inff()) ? 0.0f : v.w;
        out4[i] = v;
    }
}

extern "C" void kernel_launch(void* const* d_in, const int* in_sizes, int n_in,
                              void* d_out, int out_size, void* d_ws, size_t ws_size,
                              hipStream_t stream) {
    (void)d_ws; (void)ws_size; (void)n_in;

    const float* feat   = (const float*)d_in[0];  // [N, 64] f32
    const int*   coords = (const int*)d_in[1];    // [N, 3] i32
    float*       out    = (float*)d_out;          // [262144, 64] f32

    const int N  = in_sizes[1] / 3;               // 1,000,000
    const int n4 = out_size / 4;                  // 4,194,304 float4s

    // 1) init output to -inf (every call: idempotent, graph-replay safe)
    {
        int blocks = (n4 + 255) / 256;
        mp_init_neginf<<<blocks, 256, 0, stream>>>((float4*)out, n4);
    }
    // 2) scatter-max: one thread per (point, channel-quad)
    {
        long long total = (long long)N * NQUAD;
        int blocks = (int)((total + 255) / 256);
        mp_scatter_max<<<blocks, 256, 0, stream>>>(feat, coords, out, N);
    }
    // 3) -inf -> 0 for uncovered voxels
    {
        int blocks = (n4 + 255) / 256;
        mp_finalize<<<blocks, 256, 0, stream>>>((float4*)out, n4);
    }
}